// GCN_8967891714538
// MI455X (gfx1250) — compile-verified
//
#include <hip/hip_runtime.h>

typedef __attribute__((ext_vector_type(2))) float v2f;
typedef __attribute__((ext_vector_type(8))) float v8f;

#define GCN_N_NODES 50000
#define GCN_IN_CH   128
#define GCN_HID_CH  128
#define GCN_OUT_CH  64

// ---------------------------------------------------------------------------
// Degree / normalization kernels
// ---------------------------------------------------------------------------
__global__ void k_deg_init(float* __restrict__ deg, int n) {
  int i = blockIdx.x * blockDim.x + threadIdx.x;
  if (i < n) deg[i] = 1.0f;  // self-loop contributes 1 to every node's degree
}

__global__ void k_deg_scatter(const int* __restrict__ dst, float* __restrict__ deg, int e) {
  int i = blockIdx.x * blockDim.x + threadIdx.x;
  if (i < e) atomicAdd(&deg[dst[i]], 1.0f);
}

__global__ void k_dinv(const float* __restrict__ deg, float* __restrict__ dinv, int n) {
  int i = blockIdx.x * blockDim.x + threadIdx.x;
  if (i < n) dinv[i] = rsqrtf(deg[i]);  // deg >= 1 always (self-loops)
}

// ---------------------------------------------------------------------------
// WMMA f32 GEMM: C[M x NC] = op(A)[M x K] * B[K x NC]
//
// One wave computes one 16x16 tile of C, K-loop in steps of 4 using
// V_WMMA_F32_16X16X4_F32. NC/16 waves per block; a block covers 16 rows.
//
// The 16xK A tile is CONTIGUOUS in global memory (row-major, rows
// rowBase..rowBase+15), so it is staged once per block into LDS with the
// CDNA5 async copy path (global_load_async_to_lds_b128, ASYNCcnt), removing
// the 8x redundant per-wave A traffic. LDS rows are padded to K+4 words so
// the WMMA A-fetch (lane L -> words (K+4)*(L&15) + k + 2*(L>>4) + {0,1})
// hits banks 4m+k+{0..3}: conflict-free across the wave.
//
// A-layout (16x4 f32): lanes 0-15 -> {K=0 in v0, K=1 in v1},
//                      lanes 16-31 -> {K=2 in v0, K=3 in v1}  (ISA 7.12.2)
// B-layout (4x16 f32): row K striped across lanes; same half-wave K split.
// C/D layout: VGPR r -> rows r (lanes 0-15) and r+8 (lanes 16-31).
// ---------------------------------------------------------------------------
template <int K, int NC, bool RELU_A>
__global__ __launch_bounds__(32 * (NC / 16))
void k_gemm_wmma(const float* __restrict__ A, const float* __restrict__ B,
                 float* __restrict__ C) {
  constexpr int THREADS  = 32 * (NC / 16);
  constexpr int LDSTRIDE = K + 4;            // padded row stride (words)
  __shared__ float tileA[16 * LDSTRIDE];

  const int lane = threadIdx.x & 31;         // wave32
  const int wave = threadIdx.x >> 5;
  const int hw   = lane >> 4;                // half-wave (0 or 1)
  const int m    = lane & 15;
  const int rowBase = blockIdx.x * 16;
  const int colBase = wave * 16;

  // ---- async stage: 16 x K f32 tile, global -> LDS (b128 chunks) ----
  {
    constexpr int ROW_CHUNKS = K / 4;        // b128 chunks per row
    constexpr int CHUNKS     = 16 * ROW_CHUNKS;
    const float* gsrc = A + (size_t)rowBase * K;
    const uint32_t lds_base = (uint32_t)(uintptr_t)(&tileA[0]);
#pragma unroll
    for (int c = threadIdx.x; c < CHUNKS; c += THREADS) {
      const int row  = c / ROW_CHUNKS;
      const int col4 = c % ROW_CHUNKS;
      const uint32_t lofs = lds_base + (uint32_t)(row * LDSTRIDE + col4 * 4) * 4u;
      const uint64_t g = (uint64_t)(uintptr_t)(gsrc + (size_t)row * K + col4 * 4);
      asm volatile("global_load_async_to_lds_b128 %0, %1, off"
                   :: "v"(lofs), "v"(g) : "memory");
    }
    asm volatile("s_wait_asynccnt 0" ::: "memory");
    __syncthreads();
  }

  const float* arow = &tileA[m * LDSTRIDE];

  v8f acc = {};
#pragma unroll
  for (int k = 0; k < K; k += 4) {
    const int ka = k + 2 * hw;
    v2f a = *(const v2f*)(arow + ka);        // ds_load_b64, bank-conflict-free
    if (RELU_A) {                            // fused ReLU on layer-2 input
      a.x = fmaxf(a.x, 0.0f);
      a.y = fmaxf(a.y, 0.0f);
    }
    v2f b;
    b.x = B[(size_t)(ka + 0) * NC + colBase + m];
    b.y = B[(size_t)(ka + 1) * NC + colBase + m];
    acc = __builtin_amdgcn_wmma_f32_16x16x4_f32(
        /*neg_a=*/false, a, /*neg_b=*/false, b,
        /*c_mod=*/(short)0, acc, /*reuse_a=*/false, /*reuse_b=*/false);
  }

#pragma unroll
  for (int r = 0; r < 8; ++r) {
    const int row = rowBase + r + 8 * hw;
    C[(size_t)row * NC + colBase + m] = acc[r];
  }
}

// ---------------------------------------------------------------------------
// Aggregation init: out[n][f] = h[n][f] * dinv[n]^2 + bias[f]
// (self-loop term with norm dinv*dinv, bias folded in so edge atomics can
//  accumulate directly on top)
// ---------------------------------------------------------------------------
template <int F>
__global__ void k_self_bias(const float* __restrict__ h, const float* __restrict__ dinv,
                            const float* __restrict__ bias, float* __restrict__ out, int n) {
  constexpr int FQ = F / 4;
  int idx = blockIdx.x * blockDim.x + threadIdx.x;
  if (idx >= n * FQ) return;
  const int node = idx / FQ;
  const int f    = (idx % FQ) * 4;
  const float di = dinv[node];
  const float w  = di * di;
  const float4 hv = *(const float4*)(h + (size_t)node * F + f);
  const float4 bv = *(const float4*)(bias + f);
  float4 o;
  o.x = hv.x * w + bv.x;
  o.y = hv.y * w + bv.y;
  o.z = hv.z * w + bv.z;
  o.w = hv.w * w + bv.w;
  *(float4*)(out + (size_t)node * F + f) = o;
}

// ---------------------------------------------------------------------------
// Edge scatter: out[dst][f] += h[src][f] * dinv[src]*dinv[dst]
// One thread per (edge, 4 features): a wave covers one edge's full feature
// row (coalesced 512B / 256B gather), then 4 f32 L2 atomics per thread.
// ---------------------------------------------------------------------------
template <int F>
__global__ void k_edge_scatter(const int* __restrict__ src, const int* __restrict__ dst,
                               const float* __restrict__ h, const float* __restrict__ dinv,
                               float* __restrict__ out, int e) {
  constexpr int FQ = F / 4;
  int idx = blockIdx.x * blockDim.x + threadIdx.x;
  if (idx >= e * FQ) return;
  const int ed = idx / FQ;
  const int f  = (idx % FQ) * 4;
  const int s = src[ed];
  const int d = dst[ed];
  const float w = dinv[s] * dinv[d];
  const float4 hv = *(const float4*)(h + (size_t)s * F + f);
  float* o = out + (size_t)d * F + f;
  atomicAdd(o + 0, hv.x * w);
  atomicAdd(o + 1, hv.y * w);
  atomicAdd(o + 2, hv.z * w);
  atomicAdd(o + 3, hv.w * w);
}

// ---------------------------------------------------------------------------
// Launcher
// ---------------------------------------------------------------------------
extern "C" void kernel_launch(void* const* d_in, const int* in_sizes, int n_in,
                              void* d_out, int out_size, void* d_ws, size_t ws_size,
                              hipStream_t stream) {
  const float* x    = (const float*)d_in[0];
  const int*   eidx = (const int*)d_in[1];
  const float* W1   = (const float*)d_in[2];
  const float* b1   = (const float*)d_in[3];
  const float* W2   = (const float*)d_in[4];
  const float* b2   = (const float*)d_in[5];
  float* out = (float*)d_out;

  const int N = in_sizes[0] / GCN_IN_CH;  // 50000
  const int E = in_sizes[1] / 2;          // 800000
  const int* src = eidx;
  const int* dst = eidx + E;

  float* ws   = (float*)d_ws;
  float* deg  = ws;                                 // N
  float* dinv = deg + N;                            // N
  float* h1   = dinv + N;                           // N * HID
  float* agg1 = h1 + (size_t)N * GCN_HID_CH;        // N * HID
  float* h2   = agg1 + (size_t)N * GCN_HID_CH;      // N * OUT

  const int TB = 256;

  // --- normalization (shared by both layers) ---
  k_deg_init<<<(N + TB - 1) / TB, TB, 0, stream>>>(deg, N);
  k_deg_scatter<<<(E + TB - 1) / TB, TB, 0, stream>>>(dst, deg, E);
  k_dinv<<<(N + TB - 1) / TB, TB, 0, stream>>>(deg, dinv, N);

  // --- layer 1: h1 = x @ W1 (WMMA f32), aggregate into agg1 (bias folded) ---
  k_gemm_wmma<GCN_IN_CH, GCN_HID_CH, false>
      <<<N / 16, 32 * (GCN_HID_CH / 16), 0, stream>>>(x, W1, h1);
  {
    const int work = N * (GCN_HID_CH / 4);
    k_self_bias<GCN_HID_CH><<<(work + TB - 1) / TB, TB, 0, stream>>>(h1, dinv, b1, agg1, N);
  }
  {
    const int work = E * (GCN_HID_CH / 4);
    k_edge_scatter<GCN_HID_CH><<<(work + TB - 1) / TB, TB, 0, stream>>>(src, dst, h1, dinv, agg1, E);
  }

  // --- layer 2: h2 = relu(agg1) @ W2 (ReLU fused into LDS A-read) ---
  k_gemm_wmma<GCN_HID_CH, GCN_OUT_CH, true>
      <<<N / 16, 32 * (GCN_OUT_CH / 16), 0, stream>>>(agg1, W2, h2);
  {
    const int work = N * (GCN_OUT_CH / 4);
    k_self_bias<GCN_OUT_CH><<<(work + TB - 1) / TB, TB, 0, stream>>>(h2, dinv, b2, out, N);
  }
  {
    const int work = E * (GCN_OUT_CH / 4);
    k_edge_scatter<GCN_OUT_CH><<<(work + TB - 1) / TB, TB, 0, stream>>>(src, dst, h2, dinv, out, E);
  }
}